// SoftDtwRkdDistance_36094905156166
// MI455X (gfx1250) — compile-verified
//
#include <hip/hip_runtime.h>
#include <math.h>

typedef float v2f __attribute__((ext_vector_type(2)));
typedef float v8f __attribute__((ext_vector_type(8)));

#define TT 96
#define DDIM 128
#define LDSX 132          // 128 + 4 pad: keeps 16B alignment, kills bank conflicts
#define NPAIR 1024        // 32*32
#define INFV 1e10f

// One block per (input, a, b): blockIdx.x in [0, 2048)
//   which = bid >> 10  (0 = teacher, 1 = student)
//   p     = bid & 1023, a = p>>5, b = p&31
// Output: dtw_out[bid] = softDTW(C_{a,b})
__global__ __launch_bounds__(256)
void sdtw_fused_kernel(const float* __restrict__ student,
                       const float* __restrict__ teacher,
                       const float* __restrict__ gammaPtr,
                       float* __restrict__ dtw_out) {
    __shared__ float sA[TT * LDSX];   // 50688 B
    __shared__ float sB[TT * LDSX];   // 50688 B
    __shared__ float sC[TT * TT];     // 36864 B
    __shared__ float sXn[2 * TT];     //   768 B
    __shared__ float sR[3 * TT];      //  1152 B

    const int bid   = blockIdx.x;
    const int which = bid >> 10;               // 0: teacher, 1: student
    const int p     = bid & (NPAIR - 1);
    const int a     = p >> 5;
    const int b     = p & 31;
    const float* __restrict__ X  = (which == 0) ? teacher : student;
    const float* __restrict__ Xa = X + (size_t)a * TT * DDIM;
    const float* __restrict__ Xb = X + (size_t)b * TT * DDIM;

    const int tid = threadIdx.x;

    // ---- Stage Xa, Xb into LDS (float4, coalesced, 16B aligned both sides) ----
    for (int idx = tid * 4; idx < TT * DDIM; idx += 256 * 4) {
        int row = idx >> 7;          // / 128
        int col = idx & 127;
        float4 va = *(const float4*)(Xa + idx);
        float4 vb = *(const float4*)(Xb + idx);
        *(float4*)(&sA[row * LDSX + col]) = va;
        *(float4*)(&sB[row * LDSX + col]) = vb;
    }
    __syncthreads();

    // ---- Row squared norms ----
    if (tid < TT) {
        const float4* r = (const float4*)&sA[tid * LDSX];
        float s = 0.f;
        #pragma unroll 8
        for (int k = 0; k < DDIM / 4; ++k) {
            float4 v = r[k];
            s += v.x * v.x + v.y * v.y + v.z * v.z + v.w * v.w;
        }
        sXn[tid] = s;
    } else if (tid >= 128 && tid < 128 + TT) {
        int row = tid - 128;
        const float4* r = (const float4*)&sB[row * LDSX];
        float s = 0.f;
        #pragma unroll 8
        for (int k = 0; k < DDIM / 4; ++k) {
            float4 v = r[k];
            s += v.x * v.x + v.y * v.y + v.z * v.z + v.w * v.w;
        }
        sXn[TT + row] = s;
    }
    __syncthreads();

    // ---- GEMM: C = xn_a + xn_b - 2 * (Xa · Xb^T), via v_wmma_f32_16x16x4_f32 ----
    // 6x6 grid of 16x16 tiles, 8 waves round-robin over the 36 tiles.
    {
        const int wave = tid >> 5;
        const int lane = tid & 31;
        const int hl   = lane >> 4;     // half: 0 -> K {0,1}, 1 -> K {2,3}
        const int l16  = lane & 15;

        for (int tile = wave; tile < 36; tile += 8) {
            const int m0 = (tile / 6) * 16;
            const int n0 = (tile % 6) * 16;
            v8f acc = {0.f, 0.f, 0.f, 0.f, 0.f, 0.f, 0.f, 0.f};
            const float* ap = &sA[(m0 + l16) * LDSX + 2 * hl];
            const float* bp = &sB[(n0 + l16) * LDSX + 2 * hl];
            #pragma unroll 8
            for (int k0 = 0; k0 < DDIM; k0 += 4) {
                v2f av; av.x = ap[k0]; av.y = ap[k0 + 1];
                v2f bv; bv.x = bp[k0]; bv.y = bp[k0 + 1];
                acc = __builtin_amdgcn_wmma_f32_16x16x4_f32(
                    /*neg_a=*/false, av, /*neg_b=*/false, bv,
                    /*c_mod=*/(short)0, acc, /*reuse_a=*/false, /*reuse_b=*/false);
            }
            const float xnb = sXn[TT + n0 + l16];
            #pragma unroll
            for (int v = 0; v < 8; ++v) {
                int m = m0 + v + 8 * hl;   // C/D layout: VGPR v -> M = v + 8*half
                sC[m * TT + n0 + l16] = sXn[m] + xnb - 2.0f * acc[v];
            }
        }
    }
    __syncthreads();

    // ---- Soft-DTW wavefront DP over anti-diagonals ----
    const float gamma = gammaPtr[0];
    const float ig    = 1.0f / gamma;

    float* r2 = &sR[0];
    float* r1 = &sR[TT];
    float* rc = &sR[2 * TT];
    if (tid < TT) { r2[tid] = INFV; r1[tid] = INFV; }
    __syncthreads();

    for (int d = 0; d < 2 * TT - 1; ++d) {
        float rv = INFV;
        const int i = tid;
        if (i < TT) {
            const int j = d - i;
            if (j >= 0 && j < TT) {
                float c    = sC[i * TT + j];
                float up   = (i > 0) ? r1[i - 1] : INFV;
                float left = r1[i];
                float diag = (i > 0) ? r2[i - 1] : INFV;
                float smin;
                if (d == 0) {
                    smin = 0.0f;
                } else {
                    float m  = fminf(up, fminf(left, diag));
                    float ss = expf((m - up) * ig) + expf((m - left) * ig)
                             + expf((m - diag) * ig);
                    smin = m - gamma * logf(ss);
                }
                rv = c + smin;
                if (d == 2 * TT - 2 && i == TT - 1) dtw_out[bid] = rv;
            }
        }
        __syncthreads();                 // readers done with r1/r2
        if (i < TT) rc[i] = rv;
        __syncthreads();                 // writes visible
        float* tmp = r2; r2 = r1; r1 = rc; rc = tmp;
    }
}

// Means, normalization, smooth-L1 mean -> scalar. One block.
__global__ __launch_bounds__(256)
void finalize_kernel(const float* __restrict__ ws, float* __restrict__ out) {
    __shared__ float st[256];
    __shared__ float ss[256];
    const int tid = threadIdx.x;

    float at = 0.f, as = 0.f;
    for (int i = tid; i < NPAIR; i += 256) {
        at += ws[i];            // teacher dtw
        as += ws[NPAIR + i];    // student dtw
    }
    st[tid] = at; ss[tid] = as;
    __syncthreads();
    for (int off = 128; off > 0; off >>= 1) {
        if (tid < off) { st[tid] += st[tid + off]; ss[tid] += ss[tid + off]; }
        __syncthreads();
    }
    const float mt = st[0] * (1.0f / NPAIR);
    const float ms = ss[0] * (1.0f / NPAIR);
    __syncthreads();

    float al = 0.f;
    for (int i = tid; i < NPAIR; i += 256) {
        float dv = ws[NPAIR + i] / ms - ws[i] / mt;   // pred - target
        float ad = fabsf(dv);
        al += (ad < 1.0f) ? 0.5f * dv * dv : (ad - 0.5f);
    }
    st[tid] = al;
    __syncthreads();
    for (int off = 128; off > 0; off >>= 1) {
        if (tid < off) st[tid] += st[tid + off];
        __syncthreads();
    }
    if (tid == 0) out[0] = st[0] * (1.0f / NPAIR);
}

extern "C" void kernel_launch(void* const* d_in, const int* in_sizes, int n_in,
                              void* d_out, int out_size, void* d_ws, size_t ws_size,
                              hipStream_t stream) {
    const float* student = (const float*)d_in[0];   // [32,96,128] f32
    const float* teacher = (const float*)d_in[1];   // [32,96,128] f32
    const float* gamma   = (const float*)d_in[2];   // scalar f32
    float* ws  = (float*)d_ws;                      // 2048 floats used
    float* out = (float*)d_out;                     // 1 float

    // 2048 blocks: [0,1024) teacher pairs, [1024,2048) student pairs
    sdtw_fused_kernel<<<2 * NPAIR, 256, 0, stream>>>(student, teacher, gamma, ws);
    finalize_kernel<<<1, 256, 0, stream>>>(ws, out);
}